// GIN_53919019434437
// MI455X (gfx1250) — compile-verified
//
#include <hip/hip_runtime.h>

// ---------------------------------------------------------------------------
// GIN (3-layer) for MI455X / gfx1250.
//   per layer:  agg = segment_sum(x[src], dst);  h = relu((x + agg) @ W + b)
//   final:      log_softmax over 64 classes.
// Memory-bound on edge scatter (~2.5 GB total -> ~107us @ 23.3 TB/s);
// GEMMs (4.1 GFLOP fp32) done with V_WMMA_F32_16X16X4_F32, fused with the
// (x + agg) add, bias and ReLU so h is produced in one pass.
// ---------------------------------------------------------------------------

#define GIN_N_NODES 50000
#define GIN_D       128          // K dim (all layer inputs are 128)
#define GIN_MTILES  (GIN_N_NODES / 16)   // 3125, exact (50000 = 3125*16)
#define SH_PITCH    132          // 132 mod 64 == 4 -> conflict-free A reads

typedef float v2f __attribute__((ext_vector_type(2)));
typedef float v8f __attribute__((ext_vector_type(8)));

// ---------------------------------------------------------------- zero agg
__global__ void gin_zero_kernel(float* __restrict__ p, int n) {
    int i = blockIdx.x * blockDim.x + threadIdx.x;
    int stride = gridDim.x * blockDim.x;
    for (; i < n; i += stride) p[i] = 0.0f;
}

// ------------------------------------------------------- edge scatter-add
// One wave (32 lanes) per edge: float4 gather of the 128-float source row
// (512B fully coalesced), then 4 global_atomic_add_f32 per lane into dst.
__global__ void gin_scatter_kernel(const float* __restrict__ x,
                                   const int* __restrict__ src,
                                   const int* __restrict__ dst,
                                   float* __restrict__ agg, int E) {
    int wave = (blockIdx.x * blockDim.x + threadIdx.x) >> 5;
    int lane = threadIdx.x & 31;
    if (wave >= E) return;                       // wave-uniform exit
    int s = src[wave];
    int d = dst[wave];
    const float4* xs = (const float4*)(x + (size_t)s * GIN_D);
    float4 v = xs[lane];
    float* ad = agg + (size_t)d * GIN_D + lane * 4;
    atomicAdd(ad + 0, v.x);
    atomicAdd(ad + 1, v.y);
    atomicAdd(ad + 2, v.z);
    atomicAdd(ad + 3, v.w);
}

// ------------------------------------------------ fused (x+agg)@W + b GEMM
// One block per 16-row M-tile. Block stages the 16x128 (x+agg) tile in LDS
// (shared by all waves), then each wave computes one 16x16 output tile with
// 32 chained v_wmma_f32_16x16x4_f32 (K = 128). EXEC all-ones: no divergence.
//
// VGPR layouts (ISA 7.12.2, wave32):
//   A 16x4 f32 : lane L -> M = L&15 ; v0/v1 = K = (L>>4)*2 + {0,1}
//   B 4x16 f32 : lane L -> N = L&15 ; v0/v1 = K = (L>>4)*2 + {0,1}
//   C/D 16x16  : lane L -> N = L&15 ; VGPR r -> M = r + 8*(L>>4)
template <int N, bool RELU>
__global__ void gin_gemm_kernel(const float* __restrict__ x,
                                const float* __restrict__ agg,
                                const float* __restrict__ W,    // [128][N] row-major
                                const float* __restrict__ bias, // [N]
                                float* __restrict__ out) {      // [50000][N]
    __shared__ float sh[16 * SH_PITCH];
    const int m0  = blockIdx.x * 16;
    const int tid = threadIdx.x;
    const int NT  = N / 16;                      // waves per block

    // ---- stage h = x + agg (eps = 0 -> (1+eps)*x = x) into LDS ----
    for (int idx = tid; idx < 16 * GIN_D; idx += 32 * NT) {
        int r = idx >> 7;                        // /128
        int c = idx & 127;
        size_t g = (size_t)(m0 + r) * GIN_D + c;
        sh[r * SH_PITCH + c] = x[g] + agg[g];
    }
    __syncthreads();

    const int w    = tid >> 5;
    const int lane = tid & 31;
    const int half = lane >> 4;                  // 0: lanes 0-15, 1: 16-31
    const int mrow = lane & 15;
    const int n    = w * 16 + mrow;              // this lane's output column

    v8f acc = {0.f, 0.f, 0.f, 0.f, 0.f, 0.f, 0.f, 0.f};
    const float bv = bias[n];

#pragma unroll
    for (int kb = 0; kb < GIN_D / 4; ++kb) {
        const int k = kb * 4 + half * 2;
        // A fragment from LDS (8B-aligned: k even, pitch*4 = 528 = 0 mod 8)
        v2f a = *(const v2f*)(&sh[mrow * SH_PITCH + k]);
        v2f b;
        b.x = W[(size_t)k * N + n];
        b.y = W[(size_t)(k + 1) * N + n];
        // (neg_a, A, neg_b, B, c_mod, C, reuse_a, reuse_b)
        acc = __builtin_amdgcn_wmma_f32_16x16x4_f32(
            false, a, false, b, (short)0, acc, false, false);
    }

#pragma unroll
    for (int r = 0; r < 8; ++r) {
        float v = acc[r] + bv;
        if (RELU) v = fmaxf(v, 0.0f);
        out[(size_t)(m0 + r + 8 * half) * N + n] = v;
    }
}

// ------------------------------------------------- log_softmax over 64 cols
// One wave per node: 2 values per lane, wave32 shuffle reductions, in-place.
__global__ void gin_logsoftmax_kernel(float* __restrict__ out) {
    int node = (blockIdx.x * blockDim.x + threadIdx.x) >> 5;
    int lane = threadIdx.x & 31;
    if (node >= GIN_N_NODES) return;             // wave-uniform exit
    float* row = out + (size_t)node * 64;
    float v0 = row[lane];
    float v1 = row[lane + 32];
    float m = fmaxf(v0, v1);
#pragma unroll
    for (int off = 16; off > 0; off >>= 1)
        m = fmaxf(m, __shfl_xor(m, off, 32));
    float s = __expf(v0 - m) + __expf(v1 - m);
#pragma unroll
    for (int off = 16; off > 0; off >>= 1)
        s += __shfl_xor(s, off, 32);
    float l = m + __logf(s);
    row[lane]      = v0 - l;
    row[lane + 32] = v1 - l;
}

// ---------------------------------------------------------------------------
extern "C" void kernel_launch(void* const* d_in, const int* in_sizes, int n_in,
                              void* d_out, int out_size, void* d_ws, size_t ws_size,
                              hipStream_t stream) {
    const float* feature = (const float*)d_in[0];
    const int*   edge    = (const int*)d_in[1];   // [2, E]
    const float* W1 = (const float*)d_in[2];
    const float* b1 = (const float*)d_in[3];
    const float* Wh = (const float*)d_in[4];
    const float* bh = (const float*)d_in[5];
    const float* Wo = (const float*)d_in[6];
    const float* bo = (const float*)d_in[7];

    const int E = in_sizes[1] / 2;
    const int* src = edge;
    const int* dst = edge + E;

    // workspace: agg [50000*128] then h [50000*128]  (51.2 MB total)
    float* agg = (float*)d_ws;
    float* h   = agg + (size_t)GIN_N_NODES * GIN_D;
    float* outf = (float*)d_out;                  // [50000*64]

    const int aggN = GIN_N_NODES * GIN_D;
    const int scatterBlocks = (E + 7) / 8;        // 8 waves (edges) per block

    // ---- layer 1: agg(feature) -> h = relu((feature+agg)@W1 + b1) ----
    gin_zero_kernel<<<512, 256, 0, stream>>>(agg, aggN);
    gin_scatter_kernel<<<scatterBlocks, 256, 0, stream>>>(feature, src, dst, agg, E);
    gin_gemm_kernel<128, true><<<GIN_MTILES, 256, 0, stream>>>(feature, agg, W1, b1, h);

    // ---- layer 2: agg(h) -> h = relu((h+agg)@Wh + bh)  (in-place safe:
    //      each block LDS-loads its 16 rows before writing the same rows) ----
    gin_zero_kernel<<<512, 256, 0, stream>>>(agg, aggN);
    gin_scatter_kernel<<<scatterBlocks, 256, 0, stream>>>(h, src, dst, agg, E);
    gin_gemm_kernel<128, true><<<GIN_MTILES, 256, 0, stream>>>(h, agg, Wh, bh, h);

    // ---- layer 3: agg(h) -> logits = (h+agg)@Wo + bo -> log_softmax ----
    gin_zero_kernel<<<512, 256, 0, stream>>>(agg, aggN);
    gin_scatter_kernel<<<scatterBlocks, 256, 0, stream>>>(h, src, dst, agg, E);
    gin_gemm_kernel<64, false><<<GIN_MTILES, 128, 0, stream>>>(h, agg, Wo, bo, outf);
    gin_logsoftmax_kernel<<<(GIN_N_NODES + 7) / 8, 256, 0, stream>>>(outf);
}